// MACE_54674933678522
// MI455X (gfx1250) — compile-verified
//
#include <hip/hip_runtime.h>
#include <hip/hip_bf16.h>
#include <math.h>

#define N_NODES   16000
#define N_EDGES   256000
#define N_FEAT    64
#define N_SPECIES 10
#define N_RADIAL  8
#define EPSILON   0.25f

typedef __attribute__((ext_vector_type(2))) float v2f;
typedef __attribute__((ext_vector_type(8))) float v8f;

__device__ __forceinline__ float silu_f(float x) {
    return x / (1.0f + __expf(-x));
}

// ---------------------------------------------------------------------------
// Zero the agg accumulator (must happen every call; atomics accumulate into it)
// ---------------------------------------------------------------------------
__global__ __launch_bounds__(256) void mace_zero_kernel(float4* __restrict__ p, int n4) {
    int i = blockIdx.x * 256 + threadIdx.x;
    if (i < n4) p[i] = make_float4(0.f, 0.f, 0.f, 0.f);
}

// ---------------------------------------------------------------------------
// vskip[s,f] = sum_g W_skip[s, l=0, f, g] * W_out[g]
// Only the d=0 column of the skip connection survives into the output, and it
// folds into a per-specie 64-vector.
// ---------------------------------------------------------------------------
__global__ __launch_bounds__(64) void mace_skipvec_kernel(
    const float* __restrict__ W_skip,   // [10,4,64,64]
    const float* __restrict__ W_out,    // [64]
    float* __restrict__ vskip)          // [10,64]
{
    int s = blockIdx.x;          // 0..9
    int f = threadIdx.x;         // 0..63
    const float* W = W_skip + ((size_t)(s * 4 + 0) * 64 + f) * 64;
    float acc = 0.f;
    #pragma unroll 8
    for (int g = 0; g < 64; ++g) acc += W[g] * W_out[g];
    vskip[s * 64 + f] = acc;
}

// ---------------------------------------------------------------------------
// Fused edge kernel, 16 edges per 256-thread block (8 waves):
//   phase 1: h = silu(re @ W_rad1 + b)  -> LDS [16][64] ; Y -> LDS [16][16]
//   phase 2: R = h @ W_rad2 via V_WMMA_F32_16X16X4_F32 -> LDS [16][256]
//   phase 3: msg = Rlm * (xs + xs0*Y) * EPS, float4 gather + atomic scatter
// ---------------------------------------------------------------------------
__global__ __launch_bounds__(256) void mace_edge_kernel(
    const float* __restrict__ vectors,     // [E,3]
    const float* __restrict__ node_feats,  // [N,64,16]
    const float* __restrict__ radial,      // [E,8]
    const int*   __restrict__ senders,     // [E]
    const int*   __restrict__ receivers,   // [E]
    const float* __restrict__ W1,          // [8,64]
    const float* __restrict__ b1,          // [64]
    const float* __restrict__ W2,          // [64,256]
    float* __restrict__ agg)               // [N,64,16]
{
    __shared__ float hLDS[16][64];     // A-matrix tile (16 edges x K=64)
    __shared__ float RL[16][256];      // D tile (16 edges x 256 cols)
    __shared__ float YL[16][16];       // spherical harmonics per edge
    __shared__ int   sLDS[16];         // senders of this tile
    __shared__ int   rLDS[16];         // receivers of this tile

    const int t  = threadIdx.x;        // 0..255
    const int e0 = blockIdx.x * 16;    // 16000 blocks

    // ---- phase 1a: radial MLP layer 1 + SiLU -> hLDS -----------------------
    #pragma unroll
    for (int i = 0; i < 4; ++i) {
        int idx = t + i * 256;                 // 0..1023
        int le = idx >> 6, c = idx & 63;
        const float* re = radial + (size_t)(e0 + le) * N_RADIAL;
        float acc = b1[c];
        #pragma unroll
        for (int k = 0; k < 8; ++k) acc += re[k] * W1[k * 64 + c];
        hLDS[le][c] = silu_f(acc);
    }

    // ---- phase 1b: stage edge indices + one thread per (edge,lm) does Y ----
    if (t < 16)               sLDS[t]      = senders[e0 + t];
    else if (t < 32)          rLDS[t - 16] = receivers[e0 + t - 16];
    {
        int le = t >> 4, d = t & 15;
        const float* vv = vectors + (size_t)(e0 + le) * 3;
        float vx = vv[0], vy = vv[1], vz = vv[2];
        float r = sqrtf(vx * vx + vy * vy + vz * vz + 1e-12f);
        float x = vx / r, y = vy / r, z = vz / r;
        float x2 = x * x, y2 = y * y, z2 = z * z;
        const float s3 = 1.7320508075688772f, s15 = 3.872983346207417f;
        const float s5h = 1.118033988749895f, c358 = 2.091650066335189f;
        const float c105 = 10.246950765959598f, c218 = 1.620185174601965f;
        const float c7h = 1.3228756555322954f;
        float yv;
        switch (d) {
            case 0:  yv = 1.0f; break;
            case 1:  yv = s3 * y; break;
            case 2:  yv = s3 * z; break;
            case 3:  yv = s3 * x; break;
            case 4:  yv = s15 * x * y; break;
            case 5:  yv = s15 * y * z; break;
            case 6:  yv = s5h * (3.0f * z2 - 1.0f); break;
            case 7:  yv = s15 * x * z; break;
            case 8:  yv = 0.5f * s15 * (x2 - y2); break;
            case 9:  yv = c358 * y * (3.0f * x2 - y2); break;
            case 10: yv = c105 * x * y * z; break;
            case 11: yv = c218 * y * (5.0f * z2 - 1.0f); break;
            case 12: yv = c7h * (5.0f * z2 * z - 3.0f * z); break;
            case 13: yv = c218 * x * (5.0f * z2 - 1.0f); break;
            case 14: yv = 0.5f * c105 * z * (x2 - y2); break;
            default: yv = c358 * x * (x2 - 3.0f * y2); break;
        }
        YL[le][d] = yv;
    }
    __syncthreads();

    // ---- phase 2: R = h @ W_rad2 via WMMA f32 16x16x4 ----------------------
    // wave w handles N-tiles 2w and 2w+1 (16 cols each) of the 256 outputs.
    {
        const int wave = t >> 5;
        const int lane = t & 31;
        const int lh   = lane >> 4;     // lane half: K offset {0,2}
        const int ll   = lane & 15;     // A: row M; B/D: col N
        #pragma unroll
        for (int nti = 0; nti < 2; ++nti) {
            const int n0 = (wave * 2 + nti) * 16;
            v8f acc = {};
            #pragma unroll 4
            for (int ks = 0; ks < 16; ++ks) {
                const int kb = ks * 4 + lh * 2;
                v2f a, b;
                // A 16x4: vgpr j -> K = 4*ks + 2*lh + j, row M = ll
                a[0] = hLDS[ll][kb + 0];
                a[1] = hLDS[ll][kb + 1];
                // B 4x16: vgpr j -> K = 4*ks + 2*lh + j, col N = n0+ll
                b[0] = W2[(size_t)(kb + 0) * 256 + n0 + ll];
                b[1] = W2[(size_t)(kb + 1) * 256 + n0 + ll];
                acc = __builtin_amdgcn_wmma_f32_16x16x4_f32(
                        /*neg_a=*/false, a, /*neg_b=*/false, b,
                        /*c_mod=*/(short)0, acc,
                        /*reuse_a=*/false, /*reuse_b=*/false);
            }
            // D: vgpr r -> row M = r + 8*lh, col N = n0 + ll
            #pragma unroll
            for (int r = 0; r < 8; ++r)
                RL[r + 8 * lh][n0 + ll] = acc[r];
        }
    }
    __syncthreads();

    // ---- phase 3: message + atomic scatter (float4 gather + prefetch) ------
    #pragma unroll
    for (int i = 0; i < 4; ++i) {
        int p  = t + i * 256;            // 0..1023 -> (edge, feature)
        int le = p >> 6, f = p & 63;
        int sn = sLDS[le];
        int rc = rLDS[le];

        // prefetch the gather row for the next iteration (global_prefetch_b8)
        if (i < 3) {
            int pn = p + 256;
            int len = pn >> 6, fn = pn & 63;
            __builtin_prefetch(node_feats + ((size_t)sLDS[len] * 64 + fn) * 16, 0, 0);
        }

        const float4* xrow4 = (const float4*)(node_feats + ((size_t)sn * 64 + f) * 16);
        float4 X0 = xrow4[0], X1 = xrow4[1], X2 = xrow4[2], X3 = xrow4[3];
        float xv[16] = { X0.x, X0.y, X0.z, X0.w,  X1.x, X1.y, X1.z, X1.w,
                         X2.x, X2.y, X2.z, X2.w,  X3.x, X3.y, X3.z, X3.w };
        float xs0 = xv[0];
        float R0 = RL[le][f * 4 + 0];
        float R1 = RL[le][f * 4 + 1];
        float R2 = RL[le][f * 4 + 2];
        float R3 = RL[le][f * 4 + 3];
        float* arow = agg + ((size_t)rc * 64 + f) * 16;
        #pragma unroll
        for (int d = 0; d < 16; ++d) {
            float Rl = (d == 0) ? R0 : (d < 4) ? R1 : (d < 9) ? R2 : R3;
            float m = Rl * (xv[d] + xs0 * YL[le][d]) * EPSILON;
            atomicAdd(arow + d, m);
        }
    }
}

// ---------------------------------------------------------------------------
// Node kernel: one wave per node. gate + gated readout + skip vector dot.
// out[n] = sum_f W_out[f]*agg[n,f,0]*gate[n,f] + sum_f x0[n,f]*vskip[s,f]
// ---------------------------------------------------------------------------
__global__ __launch_bounds__(256) void mace_node_kernel(
    const float* __restrict__ agg,         // [N,64,16]
    const float* __restrict__ node_feats,  // [N,64,16]
    const int*   __restrict__ specie,      // [N]
    const float* __restrict__ c2,          // [10,64]
    const float* __restrict__ c3,          // [10,64]
    const float* __restrict__ W_out,       // [64]
    const float* __restrict__ vskip,       // [10,64]
    float* __restrict__ out)               // [N]
{
    const int wave = threadIdx.x >> 5;
    const int lane = threadIdx.x & 31;
    const int n = blockIdx.x * 8 + wave;
    if (n >= N_NODES) return;
    const int s = specie[n];
    float partial = 0.f;
    #pragma unroll
    for (int i = 0; i < 2; ++i) {
        const int f = lane + i * 32;
        const float4* arow = (const float4*)(agg + ((size_t)n * 64 + f) * 16);
        float4 A0 = arow[0], A1 = arow[1], A2 = arow[2], A3 = arow[3];
        float a0 = A0.x;
        float p2 = A0.x*A0.x + A0.y*A0.y + A0.z*A0.z + A0.w*A0.w
                 + A1.x*A1.x + A1.y*A1.y + A1.z*A1.z + A1.w*A1.w
                 + A2.x*A2.x + A2.y*A2.y + A2.z*A2.z + A2.w*A2.w
                 + A3.x*A3.x + A3.y*A3.y + A3.z*A3.z + A3.w*A3.w;
        float p3   = p2 * a0;
        float gate = 1.0f + c2[s * 64 + f] * p2 + c3[s * 64 + f] * p3;
        float x0   = node_feats[((size_t)n * 64 + f) * 16];
        partial += W_out[f] * (a0 * gate) + x0 * vskip[s * 64 + f];
    }
    #pragma unroll
    for (int off = 16; off >= 1; off >>= 1)
        partial += __shfl_xor(partial, off, 32);
    if (lane == 0) out[n] = partial;
}

// ---------------------------------------------------------------------------
extern "C" void kernel_launch(void* const* d_in, const int* in_sizes, int n_in,
                              void* d_out, int out_size, void* d_ws, size_t ws_size,
                              hipStream_t stream) {
    const float* vectors     = (const float*)d_in[0];
    const float* node_feats  = (const float*)d_in[1];
    const float* radial      = (const float*)d_in[2];
    const int*   node_specie = (const int*)  d_in[3];
    const int*   senders     = (const int*)  d_in[4];
    const int*   receivers   = (const int*)  d_in[5];
    const float* W_rad1      = (const float*)d_in[6];
    const float* b_rad1      = (const float*)d_in[7];
    const float* W_rad2      = (const float*)d_in[8];
    const float* W_skip      = (const float*)d_in[9];
    const float* c2          = (const float*)d_in[10];
    const float* c3          = (const float*)d_in[11];
    const float* W_out       = (const float*)d_in[12];
    float* out = (float*)d_out;

    // workspace: agg [N,64,16] fp32 (64 MB) + vskip [10,64]
    float* agg   = (float*)d_ws;
    float* vskip = agg + (size_t)N_NODES * 64 * 16;

    // 1) zero agg
    const int n4 = (N_NODES * 64 * 16) / 4;
    mace_zero_kernel<<<(n4 + 255) / 256, 256, 0, stream>>>((float4*)agg, n4);

    // 2) precompute skip vector
    mace_skipvec_kernel<<<N_SPECIES, 64, 0, stream>>>(W_skip, W_out, vskip);

    // 3) fused edge kernel (16 edges per block)
    mace_edge_kernel<<<N_EDGES / 16, 256, 0, stream>>>(
        vectors, node_feats, radial, senders, receivers,
        W_rad1, b_rad1, W_rad2, agg);

    // 4) node gate + readout (8 nodes per block, one wave each)
    mace_node_kernel<<<(N_NODES + 7) / 8, 256, 0, stream>>>(
        agg, node_feats, node_specie, c2, c3, W_out, vskip, out);
}